// ExpKernelMVHawkesProcess_64244120814126
// MI455X (gfx1250) — compile-verified
//
#include <hip/hip_runtime.h>

typedef __attribute__((ext_vector_type(2))) float v2f;
typedef __attribute__((ext_vector_type(8))) float v8f;

#define B_ 4
#define N_ 2048
#define D_ 32
#define WAVES_PER_BLOCK 8
#define TILE_ROWS 16

// workspace layout (float offsets)
#define WS_MU     0
#define WS_ALPHA  (WS_MU + D_)
#define WS_BETA   (WS_ALPHA + D_*D_)
#define WS_AB     (WS_BETA + D_*D_)
#define WS_SUMMU  (WS_AB + D_*D_)
#define WS_ROW    (WS_SUMMU + 32)          /* padded; B_*N_ floats of per-row values */

__device__ __forceinline__ float softplus_f(float x) {
    // stable: max(x,0) + log1p(exp(-|x|)) == jax.nn.softplus in fp32
    return fmaxf(x, 0.0f) + log1pf(expf(-fabsf(x)));
}

// CDNA5 async memory->LDS copy (16B per lane). VDST operand supplies the LDS
// byte address; generic LDS pointers carry the LDS offset in addr[31:0]
// (ISA 10.2 aperture rules), so truncation yields a valid DS address.
__device__ __forceinline__ void async_copy_b128(const float* gptr, float* lptr) {
    unsigned lds            = (unsigned)(unsigned long long)lptr;
    unsigned long long gadd = (unsigned long long)gptr;
    asm volatile("global_load_async_to_lds_b128 %0, %1, off"
                 :: "v"(lds), "v"(gadd) : "memory");
}
__device__ __forceinline__ void async_wait0() {
    asm volatile("s_wait_asynccnt 0x0" ::: "memory");
}

// ---------------- Kernel 1: parameter transform ----------------
__global__ void hawkes_params(const float* __restrict__ mu_raw,
                              const float* __restrict__ alpha_raw,
                              const float* __restrict__ beta_raw,
                              float* __restrict__ ws) {
    __shared__ float smu[D_];
    const int tid = threadIdx.x;
    if (tid < D_) {
        float m = softplus_f(mu_raw[tid]);
        ws[WS_MU + tid] = m;
        smu[tid] = m;
    }
    if (tid < D_ * D_) {
        float a = softplus_f(alpha_raw[tid]);
        float b = softplus_f(beta_raw[tid]);
        ws[WS_ALPHA + tid] = a;
        ws[WS_BETA  + tid] = b;
        ws[WS_AB    + tid] = a * b;
    }
    __syncthreads();
    if (tid == 0) {
        float s = 0.0f;
        for (int d = 0; d < D_; ++d) s += smu[d];   // fixed order: deterministic
        ws[WS_SUMMU] = s;
    }
}

// ---------------- Kernel 2: pairwise tiles + compensator ----------------
__global__ void __launch_bounds__(256)
hawkes_main(const float* __restrict__ tp, const int* __restrict__ et,
            const float* __restrict__ Tg, const float* __restrict__ ws,
            float* __restrict__ rowv) {
    __shared__ float s_t[N_];
    __shared__ int   s_e[N_];
    __shared__ float s_alpha[D_ * D_];
    __shared__ float s_beta[D_ * D_];
    __shared__ float s_ab[D_ * D_];
    __shared__ float s_mu[D_];
    __shared__ float s_dump[WAVES_PER_BLOCK * 256];   // 8 VGPRs x 32 lanes per wave

    const int b   = blockIdx.y;
    const int tid = threadIdx.x;

    // --- Async staging: memory -> LDS via ASYNCcnt-tracked DMA (no VGPR bounce).
    // Per-batch series: 2048 f32 = 512 x 16B chunks each.
    {
        const float* tsrc = tp + b * N_;
        const float* esrc = (const float*)(et + b * N_);
#pragma unroll
        for (int k = 0; k < 2; ++k) {
            const int c = tid + k * 256;              // chunk id, 512 chunks
            async_copy_b128(tsrc + c * 4, &s_t[c * 4]);
            async_copy_b128(esrc + c * 4, (float*)&s_e[c * 4]);
        }
        // Tables: 1024 f32 = 256 chunks each (one per thread).
        async_copy_b128(ws + WS_ALPHA + tid * 4, &s_alpha[tid * 4]);
        async_copy_b128(ws + WS_BETA  + tid * 4, &s_beta[tid * 4]);
        async_copy_b128(ws + WS_AB    + tid * 4, &s_ab[tid * 4]);
        if (tid < D_ / 4) async_copy_b128(ws + WS_MU + tid * 4, &s_mu[tid * 4]);
    }
    async_wait0();
    __syncthreads();

    const int wave = tid >> 5;
    const int lane = tid & 31;
    const int tile = blockIdx.x * WAVES_PER_BLOCK + wave;   // 0..127
    const int i0   = tile * TILE_ROWS;
    const int m    = lane & 15;       // row within tile (A-matrix M index)
    const int half = lane >> 4;       // selects K=0,1 vs K=2,3 per ISA A-layout
    const int irow = i0 + m;

    const float ti      = s_t[irow];
    const int   rowbase = s_e[irow] << 5;   // alpha row e_i * D

    v8f c = {0.f, 0.f, 0.f, 0.f, 0.f, 0.f, 0.f, 0.f};
    v2f onesb; onesb.x = 1.0f; onesb.y = 1.0f;    // B = 4x16 all-ones

    // Trip count is wave-uniform; make it scalar so the backend emits an
    // s_cmp/s_cbranch loop instead of the exec-mask divergent-loop idiom.
    const int jend = __builtin_amdgcn_readfirstlane(i0 + TILE_ROWS);
    for (int j0 = 0; j0 < jend; j0 += 4) {
        const int ja = j0 + (half << 1);
        v2f av;
        {
            const int jj  = ja;
            const float tj = s_t[jj];
            const int  idx = rowbase + s_e[jj];
            const float ex = fmaxf(-s_beta[idx] * (ti - tj), -20.0f);
            const float v  = s_ab[idx] * __expf(ex);
            av.x = (jj < irow) ? v : 0.0f;          // strict lower triangle, branchless
        }
        {
            const int jj  = ja + 1;
            const float tj = s_t[jj];
            const int  idx = rowbase + s_e[jj];
            const float ex = fmaxf(-s_beta[idx] * (ti - tj), -20.0f);
            const float v  = s_ab[idx] * __expf(ex);
            av.y = (jj < irow) ? v : 0.0f;
        }
        // C += P(16x4) x Ones(4x16): matrix pipe does row-sum + accumulate,
        // co-executing with the VALU/TRANS exp work.
        c = __builtin_amdgcn_wmma_f32_16x16x4_f32(
                /*neg_a=*/false, av, /*neg_b=*/false, onesb,
                /*c_mod=*/(short)0, c, /*reuse_a=*/false, /*reuse_b=*/false);
    }

    // Dump accumulator: every column of C holds the row sum.
    {
        float* dp = &s_dump[wave * 256 + lane * 8];
#pragma unroll
        for (int v = 0; v < 8; ++v) dp[v] = c[v];
    }
    __syncthreads();

    // Compensator: lane = d (D_ == wave32 width), deterministic xor-tree reduce.
    const float Tb = Tg[b];
    for (int r = 0; r < TILE_ROWS; ++r) {
        const int   ir    = i0 + r;
        const int   er    = s_e[ir];
        const float delta = Tb - s_t[ir];
        const int   idx   = (lane << 5) + er;       // alpha[d=lane, e_i]
        float term = s_alpha[idx] * (1.0f - __expf(-s_beta[idx] * delta));
#pragma unroll
        for (int off = 16; off; off >>= 1) term += __shfl_xor(term, off, 32);
        if (lane == 0) {
            // C layout: rows 0-7 live in lane 0 (N=0), rows 8-15 in lane 16.
            const float S = (r < 8) ? s_dump[wave * 256 + r]
                                    : s_dump[wave * 256 + 128 + (r - 8)];
            const float inten = s_mu[er] + S;
            rowv[b * N_ + ir] = logf(fmaxf(inten, 1e-12f)) - term;
        }
    }
}

// ---------------- Kernel 3: deterministic per-batch reduction ----------------
__global__ void hawkes_reduce(const float* __restrict__ rowv,
                              const float* __restrict__ Tg,
                              const float* __restrict__ ws,
                              float* __restrict__ out) {
    __shared__ float red[256];
    const int b = blockIdx.x, tid = threadIdx.x;
    float s = 0.0f;
    for (int i = tid; i < N_; i += 256) s += rowv[b * N_ + i];  // fixed per-thread order
    red[tid] = s;
    __syncthreads();
    for (int off = 128; off; off >>= 1) {
        if (tid < off) red[tid] += red[tid + off];
        __syncthreads();
    }
    if (tid == 0) out[b] = red[0] - Tg[b] * ws[WS_SUMMU];
}

extern "C" void kernel_launch(void* const* d_in, const int* in_sizes, int n_in,
                              void* d_out, int out_size, void* d_ws, size_t ws_size,
                              hipStream_t stream) {
    const float* tp        = (const float*)d_in[0];  // time_points (B,N)
    const float* Tg        = (const float*)d_in[1];  // T (B,)
    const float* mu_raw    = (const float*)d_in[2];  // (D,)
    const float* alpha_raw = (const float*)d_in[3];  // (D,D)
    const float* beta_raw  = (const float*)d_in[4];  // (D,D)
    const int*   et        = (const int*)d_in[5];    // event_types (B,N)
    float* ws  = (float*)d_ws;
    float* out = (float*)d_out;

    hawkes_params<<<1, 1024, 0, stream>>>(mu_raw, alpha_raw, beta_raw, ws);

    dim3 grid(N_ / TILE_ROWS / WAVES_PER_BLOCK, B_);   // (16, 4)
    hawkes_main<<<grid, 256, 0, stream>>>(tp, et, Tg, ws, ws + WS_ROW);

    hawkes_reduce<<<B_, 256, 0, stream>>>(ws + WS_ROW, Tg, ws, out);
}